// CylindricalSchrodingerSolver_69595650065021
// MI455X (gfx1250) — compile-verified
//
#include <hip/hip_runtime.h>

// CDNA5 / gfx1250: wave32, WMMA f32 16x16x4.
typedef __attribute__((ext_vector_type(2))) float v2f;
typedef __attribute__((ext_vector_type(8))) float v8f;

#define KDIM   812               // 28*29 grid points
#define KPAD   816               // 51 * 16
#define NT     51                // 16-row tiles
#define BATCH  8
#define NALPHA 5
#define NCOL   16                // block-power-iteration width (WMMA N)
#define NITER  4000              // multiple of NORM_EVERY
#define NORM_EVERY 8             // ||B||inf <= ~53; 53^8 ~ 6e13 << f32 max
#define CSHIFT 48.0f             // > Gershgorin bound of H => B = cI - H is nonnegative
#define TPB    1024
#define NWAVES (TPB / 32)
#define TPW    2                 // tiles per wave (NWAVES*TPW = 64 >= NT)
#define NSTEP  20                // FIXED WMMA k-steps per tile (80-col band window)
#define KBMAX  (KPAD - 4 * NSTEP) // 736: window [kb, kb+80) always inside [0, KPAD)
#define XSZ    (KPAD * NCOL)     // one X buffer, in floats

__global__ void cyl_init_ws(unsigned int* __restrict__ ws_keys) {
    int i = threadIdx.x;
    if (i < BATCH) ws_keys[i] = 0xFFFFFFFFu;   // +inf under the monotone key map
}

// Branch-free banded tile GEMM: (L X)_tile via 20 chained f32 WMMAs,
// A register-resident, B from LDS. Two accumulators break the D->C chain.
__device__ __forceinline__ v8f band_gemm(const float* __restrict__ Xc,
                                         const v2f (&af)[NSTEP], int x0) {
    v8f c0 = {0.f,0.f,0.f,0.f,0.f,0.f,0.f,0.f};
    v8f c1 = {0.f,0.f,0.f,0.f,0.f,0.f,0.f,0.f};
    #pragma unroll
    for (int s = 0; s < NSTEP; ++s) {
        v2f bv;
        bv.x = Xc[x0 + s * (4 * NCOL)];
        bv.y = Xc[x0 + s * (4 * NCOL) + NCOL];
        if (s & 1)
            c1 = __builtin_amdgcn_wmma_f32_16x16x4_f32(
                    false, af[s], false, bv, (short)0, c1, false, false);
        else
            c0 = __builtin_amdgcn_wmma_f32_16x16x4_f32(
                    false, af[s], false, bv, (short)0, c0, false, false);
    }
    return c0 + c1;
}

// One block per (alpha, batch) matrix. Shifted block power iteration:
//   Z = (cI - H) X = (c - V) .* X  -  L X
//   X <- Z (ping-pong LDS buffers), rescaled by 1/||Z||inf every 8th iter.
// L is banded (offsets {0,+-1,+-29}); each 16-row tile uses a FIXED 80-column
// window [kb, kb+80), kb = clamp(16t-32, 0, 736); out-of-band entries are
// exact zeros, cols >= 812 / dummy tiles get zero A-fragments.
__global__ __launch_bounds__(TPB, 1) void cyl_powiter(
    const float* __restrict__ mean_fields,   // [BATCH, KDIM]
    const float* __restrict__ lap,           // [KDIM, KDIM] row-major
    const float* __restrict__ inv_rho,       // [KDIM]
    unsigned int* __restrict__ ws_keys)      // [BATCH] monotone float keys
{
    extern __shared__ float smem[];          // 108096 B dynamic LDS
    float* Xbuf   = smem;                    // [2][XSZ] ping-pong
    float* Vs     = smem + 2 * XSZ;          // [KPAD]
    float* redmax = Vs + KPAD;               // [NWAVES]
    float* rednum = redmax + NWAVES;         // [NWAVES]
    float* redden = rednum + NWAVES;         // [NWAVES]

    const int tid   = threadIdx.x;
    const int wave  = tid >> 5;
    const int lane  = tid & 31;
    const int nidx  = lane & 15;             // column index n
    const int khalf = lane >> 4;             // lane-half (selects K pair / M half)

    const int a = blockIdx.x / BATCH;        // alpha
    const int b = blockIdx.x % BATCH;        // batch

    // ---- effective diagonal potential V(r) = v[b,r] + (alpha/rho)^2 ----
    for (int r = tid; r < KPAD; r += TPB) {
        float val = 0.0f;
        if (r < KDIM) {
            float t = (float)a * inv_rho[r];
            val = mean_fields[b * KDIM + r] + t * t;
        }
        Vs[r] = val;
    }
    // ---- positive deterministic start vectors (Perron eigvec is positive) ----
    for (int i = tid; i < XSZ; i += TPB) {
        int   r   = i >> 4;
        float val = 0.0f;
        if (r < KDIM) {
            unsigned h = (unsigned)i * 2654435761u;
            val = 1.0f + (float)(h >> 27) * 0.03125f;   // in [1, 2)
        }
        Xbuf[i] = val;                                  // buffer 0
    }

    // ---- preload banded A-fragments into registers (persist across iters) ----
    // A 16x4 f32 layout: lane holds row (t*16 + nidx), K pair 2*khalf{+0,+1}.
    v2f afrag[TPW][NSTEP];
    int x0a[TPW];
    #pragma unroll
    for (int ti = 0; ti < TPW; ++ti) {
        const int t = wave * TPW + ti;
        int kb = t * 16 - 32;
        kb = kb < 0 ? 0 : (kb > KBMAX ? KBMAX : kb);
        x0a[ti] = (kb + 2 * khalf) * NCOL + nidx;       // LDS elem idx of B[kb+2h][n]
        const int row  = t * 16 + nidx;
        const int rowc = row < KDIM ? row : (KDIM - 1);
        const float* Lrow = lap + (size_t)rowc * KDIM;
        #pragma unroll
        for (int s = 0; s < NSTEP; ++s) {
            const int  c  = kb + 4 * s + 2 * khalf;     // even
            const bool ok = (t < NT) && (c + 1 < KDIM); // one-time divergence OK
            v2f av = {0.f, 0.f};
            if (ok) {
                float2 l2 = *reinterpret_cast<const float2*>(Lrow + c); // 8B aligned
                av.x = l2.x; av.y = l2.y;
            }
            afrag[ti][s] = av;
        }
    }
    __syncthreads();

    int cur = 0;
    for (int it = 0; it < NITER; ++it) {
        const float* Xc = Xbuf + cur * XSZ;
        float*       Xn = Xbuf + (cur ^ 1) * XSZ;

        if ((it & (NORM_EVERY - 1)) == (NORM_EVERY - 1)) {
            // ---- normalizing iteration: needs block max => extra barrier ----
            v8f   acc[TPW];
            float zmax_l = 0.0f;
            #pragma unroll
            for (int ti = 0; ti < TPW; ++ti) {
                const int t = wave * TPW + ti;
                v8f c = band_gemm(Xc, afrag[ti], x0a[ti]);
                #pragma unroll
                for (int g = 0; g < 8; ++g) {
                    const int  r  = t * 16 + g + 8 * khalf;
                    const int  rc = r < KPAD ? r : (KPAD - 1);
                    const bool ok = (t < NT) && (r < KDIM);
                    float x = Xc[rc * NCOL + nidx];
                    float z = (CSHIFT - Vs[rc]) * x - c[g];
                    z = ok ? z : 0.0f;
                    c[g] = z;
                    float az = fabsf(z);
                    zmax_l = az > zmax_l ? az : zmax_l;
                }
                acc[ti] = c;
            }
            #pragma unroll
            for (int off = 16; off > 0; off >>= 1)
                zmax_l = fmaxf(zmax_l, __shfl_xor(zmax_l, off, 32));
            if (lane == 0) redmax[wave] = zmax_l;
            __syncthreads();
            float gmax = redmax[0];
            #pragma unroll
            for (int wv = 1; wv < NWAVES; ++wv) gmax = fmaxf(gmax, redmax[wv]);
            const float scale = 1.0f / gmax;
            #pragma unroll
            for (int ti = 0; ti < TPW; ++ti) {
                const int t = wave * TPW + ti;
                if (t < NT) {
                    #pragma unroll
                    for (int g = 0; g < 8; ++g) {
                        const int r = t * 16 + g + 8 * khalf;
                        Xn[r * NCOL + nidx] = acc[ti][g] * scale;
                    }
                }
            }
        } else {
            // ---- plain iteration: write-through, single barrier ----
            #pragma unroll
            for (int ti = 0; ti < TPW; ++ti) {
                const int t = wave * TPW + ti;
                v8f c = band_gemm(Xc, afrag[ti], x0a[ti]);
                if (t < NT) {
                    #pragma unroll
                    for (int g = 0; g < 8; ++g) {
                        const int  r  = t * 16 + g + 8 * khalf;
                        const int  rc = r < KPAD ? r : (KPAD - 1);
                        float x = Xc[rc * NCOL + nidx];
                        float z = (CSHIFT - Vs[rc]) * x - c[g];
                        z = (r < KDIM) ? z : 0.0f;
                        Xn[r * NCOL + nidx] = z;
                    }
                }
            }
        }
        __syncthreads();   // Xn complete & all Xc reads done before next flip
        cur ^= 1;
    }

    // ---- epilogue: Rayleigh quotient tr(X^T B X)/tr(X^T X) on final X ----
    {
        const float* Xc = Xbuf + cur * XSZ;
        float num = 0.0f, den = 0.0f;
        #pragma unroll
        for (int ti = 0; ti < TPW; ++ti) {
            const int t = wave * TPW + ti;
            v8f c = band_gemm(Xc, afrag[ti], x0a[ti]);
            #pragma unroll
            for (int g = 0; g < 8; ++g) {
                const int  r  = t * 16 + g + 8 * khalf;
                const int  rc = r < KPAD ? r : (KPAD - 1);
                const bool ok = (t < NT) && (r < KDIM);
                float x = Xc[rc * NCOL + nidx];
                float z = (CSHIFT - Vs[rc]) * x - c[g];
                x = ok ? x : 0.0f;
                z = ok ? z : 0.0f;
                num += x * z;
                den += x * x;
            }
        }
        #pragma unroll
        for (int off = 16; off > 0; off >>= 1) {
            num += __shfl_xor(num, off, 32);
            den += __shfl_xor(den, off, 32);
        }
        if (lane == 0) { rednum[wave] = num; redden[wave] = den; }
        __syncthreads();
        if (tid == 0) {
            float tnum = 0.f, tden = 0.f;
            for (int wv = 0; wv < NWAVES; ++wv) { tnum += rednum[wv]; tden += redden[wv]; }
            float lam_min = CSHIFT - tnum / tden;   // c - lambda_max(B)
            unsigned bits = __float_as_uint(lam_min);
            unsigned key  = (bits & 0x80000000u) ? ~bits : (bits | 0x80000000u);
            atomicMin(&ws_keys[b], key);
        }
    }
}

__global__ void cyl_finalize(const int* __restrict__ num_nucleons,
                             const unsigned int* __restrict__ ws_keys,
                             float* __restrict__ out)
{
    int b = threadIdx.x;
    if (b < BATCH) {
        unsigned key  = ws_keys[b];
        unsigned bits = (key & 0x80000000u) ? (key & 0x7FFFFFFFu) : ~key;
        float lam = __uint_as_float(bits);
        int n = num_nucleons[b];
        n = n < 0 ? 0 : (n > 2 ? 2 : n);
        out[b] = lam * (float)n;
    }
}

extern "C" void kernel_launch(void* const* d_in, const int* in_sizes, int n_in,
                              void* d_out, int out_size, void* d_ws, size_t ws_size,
                              hipStream_t stream) {
    (void)in_sizes; (void)n_in; (void)out_size; (void)ws_size;
    const int*   nn      = (const int*)d_in[0];     // [8,1] int32
    const float* mf      = (const float*)d_in[1];   // [8,28,29] f32
    const float* lap     = (const float*)d_in[2];   // [812,812] f32
    const float* inv_rho = (const float*)d_in[3];   // [812] f32
    // d_in[4] = max_alpha (==4, baked into NALPHA)
    unsigned int* keys = (unsigned int*)d_ws;

    const size_t smem_bytes = (size_t)(2 * XSZ + KPAD + 3 * NWAVES) * sizeof(float);
    cyl_init_ws<<<1, 32, 0, stream>>>(keys);
    cyl_powiter<<<NALPHA * BATCH, TPB, smem_bytes, stream>>>(mf, lap, inv_rho, keys);
    cyl_finalize<<<1, 32, 0, stream>>>(nn, keys, (float*)d_out);
}